// CostVolume_54778012893413
// MI455X (gfx1250) — compile-verified
//
#include <hip/hip_runtime.h>

typedef float v2f __attribute__((ext_vector_type(2)));
typedef float v4f __attribute__((ext_vector_type(4)));
typedef float v8f __attribute__((ext_vector_type(8)));

#define MD_   4
#define B_    4
#define C_    256
#define H_    64
#define W_    64
// output: (B, H*W, H, W) fp32

// ---------------------------------------------------------------------------
// Kernel 1: zero-fill the 256MB output with b128 stores (bandwidth-bound part)
// ---------------------------------------------------------------------------
__global__ void cv_zero_kernel(v4f* __restrict__ out, long n4) {
    long stride = (long)gridDim.x * blockDim.x;
    v4f z = {0.f, 0.f, 0.f, 0.f};
    for (long i = (long)blockIdx.x * blockDim.x + threadIdx.x; i < n4; i += stride)
        out[i] = z;
}

// ---------------------------------------------------------------------------
// Kernel 2: banded GEMM via V_WMMA_F32_16X16X4_F32 + predicated scatter.
// One wave handles (b, y, x-tile of 16, di). A = feat1[b,:,y,x0:x0+16] (16x256),
// B = feat2[b,:,y2,u] over u in [x0-4, x0+28) as two 16-col N tiles.
// cv value for (x=x0+m, dj) lives at D[m][n] with n = m + dj (band).
// ---------------------------------------------------------------------------
__global__ __launch_bounds__(256)
void cv_corr_wmma_kernel(const float* __restrict__ f1,
                         const float* __restrict__ f2,
                         float* __restrict__ out) {
    const int lane = threadIdx.x & 31;
    const int wave = threadIdx.x >> 5;
    const int unit = blockIdx.x * 8 + wave;          // 9216 units total

    const int di   = unit % 9;
    const int tile = (unit / 9) & 3;
    const int y    = (unit / 36) % H_;
    const int b    = unit / (H_ * 4 * 9);

    const int y2 = y + di - MD_;
    if ((unsigned)y2 >= H_) return;                  // wave-uniform: whole di row OOB

    const int x0 = tile * 16;
    const int n  = lane & 15;                        // A row m-part / B col within 16
    const int kk = (lane >> 4) * 2;                  // K sub-offset: lanes 16-31 hold K+2

    // A frag source: feat1[b][kk + 4s (+1)][y][x0+n]  (coalesced over lanes 0-15 / 16-31)
    const float* aptr = f1 + (((long)(b * C_ + kk) * H_ + y) * W_ + x0 + n);

    // B frag sources: columns u0 = x0-4+n (tile0), u1 = x0+12+n (tile1); clamp OOB, zero later
    const int  u0 = x0 - 4 + n;
    const int  u1 = x0 + 12 + n;
    const bool v0 = (unsigned)u0 < W_;
    const bool v1 = (unsigned)u1 < W_;
    const float* bptr0 = f2 + (((long)(b * C_ + kk) * H_ + y2) * W_ + (v0 ? u0 : 0));
    const float* bptr1 = f2 + (((long)(b * C_ + kk) * H_ + y2) * W_ + (v1 ? u1 : 0));

    const long cstep = (long)4 * H_ * W_;            // advance 4 channels per K-step
    const long c1    = (long)H_ * W_;                // +1 channel

    v8f acc0 = {0.f, 0.f, 0.f, 0.f, 0.f, 0.f, 0.f, 0.f};
    v8f acc1 = {0.f, 0.f, 0.f, 0.f, 0.f, 0.f, 0.f, 0.f};

    for (int s = 0; s < C_ / 4; ++s) {
        v2f a, bv0, bv1;
        a.x = aptr[0];
        a.y = aptr[c1];
        float t0a = bptr0[0], t0b = bptr0[c1];
        float t1a = bptr1[0], t1b = bptr1[c1];
        bv0.x = v0 ? t0a : 0.f;  bv0.y = v0 ? t0b : 0.f;   // cndmask, EXEC stays all-1s
        bv1.x = v1 ? t1a : 0.f;  bv1.y = v1 ? t1b : 0.f;
        acc0 = __builtin_amdgcn_wmma_f32_16x16x4_f32(false, a, false, bv0,
                                                     (short)0, acc0, false, false);
        acc1 = __builtin_amdgcn_wmma_f32_16x16x4_f32(false, a, false, bv1,
                                                     (short)0, acc1, false, false);
        aptr  += cstep;
        bptr0 += cstep;
        bptr1 += cstep;
    }

    // Scatter. C/D layout: VGPR j, lanes 0-15 -> M=j, lanes 16-31 -> M=j+8; N = lane&15.
    // tile0: tc = u0, dj = n - m         (valid iff 0 <= n-m <= 8 and v0)
    // tile1: tc = u1, dj = n - m + 16    (valid iff 0 <= n-m+16 <= 8 and v1)
    const int mhi = (lane >> 4) * 8;
    #pragma unroll
    for (int j = 0; j < 8; ++j) {
        const int m = j + mhi;
        const int x = x0 + m;
        const int d0 = n - m;            // dj for tile0
        const int d1 = n - m + 16;       // dj for tile1
        if (v0 && (unsigned)d0 <= 8u) {
            long idx = (((long)b * (H_ * W_) + (long)y2 * W_ + u0) * H_ + y) * W_ + x;
            out[idx] = acc0[j];
        }
        if (v1 && (unsigned)d1 <= 8u) {
            long idx = (((long)b * (H_ * W_) + (long)y2 * W_ + u1) * H_ + y) * W_ + x;
            out[idx] = acc1[j];
        }
    }
}

// ---------------------------------------------------------------------------
extern "C" void kernel_launch(void* const* d_in, const int* in_sizes, int n_in,
                              void* d_out, int out_size, void* d_ws, size_t ws_size,
                              hipStream_t stream) {
    const float* f1 = (const float*)d_in[0];
    const float* f2 = (const float*)d_in[1];
    float* out = (float*)d_out;

    // 1) zero-fill output (out_size = 4*4096*64*64, divisible by 4)
    long n4 = (long)out_size / 4;
    cv_zero_kernel<<<4096, 256, 0, stream>>>((v4f*)out, n4);

    // 2) WMMA correlation + scatter: 4*64*4*9 = 9216 waves, 8 waves/block
    cv_corr_wmma_kernel<<<9216 / 8, 256, 0, stream>>>(f1, f2, out);
}